// Generator_40673340293385
// MI455X (gfx1250) — compile-verified
//
#include <hip/hip_runtime.h>
#include <hip/hip_bf16.h>
#include <math.h>

typedef __attribute__((ext_vector_type(2))) float v2f;
typedef __attribute__((ext_vector_type(8))) float v8f;

// ---------------------------------------------------------------------------
// Pre-kernel: fold the entire gate sequence into two 10-term real quadratic
// forms (one per output Z expectation). Runs once, 1 active thread, writes
// 2 x 12 floats (K padded to 12) into d_ws.
// ---------------------------------------------------------------------------
struct Cx { float r, i; };
__device__ __forceinline__ Cx cmul(Cx a, Cx b) { return Cx{a.r*b.r - a.i*b.i, a.r*b.i + a.i*b.r}; }
__device__ __forceinline__ Cx cconjf(Cx a) { return Cx{a.r, -a.i}; }

__global__ void precompute_weights(const float* __restrict__ qw, float* __restrict__ ws) {
    if (threadIdx.x != 0 || blockIdx.x != 0) return;

    Cx U[4][4];
    for (int i = 0; i < 4; ++i)
        for (int j = 0; j < 4; ++j) U[i][j] = Cx{(i == j) ? 1.f : 0.f, 0.f};

    for (int l = 0; l < 2; ++l) {
        // Rot gates on wires 0,1:  RZ(omega) RY(theta) RZ(phi)
        Cx G[2][2][2];
        for (int w = 0; w < 2; ++w) {
            float phi = qw[l*6 + w*3 + 0];
            float th  = qw[l*6 + w*3 + 1];
            float om  = qw[l*6 + w*3 + 2];
            float ct = cosf(0.5f*th), st = sinf(0.5f*th);
            float po = 0.5f*(phi + om), mo = 0.5f*(phi - om);
            Cx epo = Cx{cosf(po), -sinf(po)};   // e^{-i(phi+om)/2}
            Cx emo = Cx{cosf(mo),  sinf(mo)};   // e^{+i(phi-om)/2}
            G[w][0][0] = Cx{ epo.r*ct,  epo.i*ct};
            G[w][0][1] = Cx{-emo.r*st, -emo.i*st};
            G[w][1][0] = Cx{ emo.r*st, -emo.i*st};  // conj(emo)*st
            G[w][1][1] = Cx{ epo.r*ct, -epo.i*ct};  // conj(epo)*ct
        }
        // M = (G0 (x) I) * U
        Cx M[4][4];
        for (int i = 0; i < 4; ++i) {
            int a = i >> 1, b = i & 1;
            for (int j = 0; j < 4; ++j) {
                Cx acc{0, 0};
                for (int c = 0; c < 2; ++c) {
                    Cx t = cmul(G[0][a][c], U[2*c + b][j]);
                    acc.r += t.r; acc.i += t.i;
                }
                M[i][j] = acc;
            }
        }
        // M2 = (I (x) G1) * M
        Cx M2[4][4];
        for (int i = 0; i < 4; ++i) {
            int a = i >> 1, b = i & 1;
            for (int j = 0; j < 4; ++j) {
                Cx acc{0, 0};
                for (int d = 0; d < 2; ++d) {
                    Cx t = cmul(G[1][b][d], M[2*a + d][j]);
                    acc.r += t.r; acc.i += t.i;
                }
                M2[i][j] = acc;
            }
        }
        // CNOT(0->1): new[i] = old[s01[i]] ; CNOT(1->0): new[i] = old[s10[i]]
        const int s01[4] = {0, 1, 3, 2};
        const int s10[4] = {0, 3, 2, 1};
        Cx M3[4][4];
        for (int i = 0; i < 4; ++i)
            for (int j = 0; j < 4; ++j) M3[i][j] = M2[s01[i]][j];
        for (int i = 0; i < 4; ++i)
            for (int j = 0; j < 4; ++j) U[i][j] = M3[s10[i]][j];
    }

    // Embedding phase diag: d = (1, -i, -i, -1)
    const float dph_r[4] = {1.f, 0.f, 0.f, -1.f};
    const float dph_i[4] = {0.f, -1.f, -1.f, 0.f};

    for (int jz = 0; jz < 2; ++jz) {
        float dz[4];
        if (jz == 0) { dz[0]=1.f; dz[1]=1.f;  dz[2]=-1.f; dz[3]=-1.f; }  // Z on wire 0
        else         { dz[0]=1.f; dz[1]=-1.f; dz[2]=1.f;  dz[3]=-1.f; }  // Z on wire 1
        // Mh = U^dag Dz U (Hermitian)
        Cx Mh[4][4];
        for (int p = 0; p < 4; ++p)
            for (int q = 0; q < 4; ++q) {
                Cx acc{0, 0};
                for (int k = 0; k < 4; ++k) {
                    Cx t = cmul(cconjf(U[k][p]), U[k][q]);
                    acc.r += dz[k]*t.r; acc.i += dz[k]*t.i;
                }
                Mh[p][q] = acc;
            }
        // S = Re( conj(d_p) * Mh * d_q )
        float S[4][4];
        for (int p = 0; p < 4; ++p)
            for (int q = 0; q < 4; ++q) {
                Cx dp = Cx{dph_r[p], -dph_i[p]};
                Cx dq = Cx{dph_r[q],  dph_i[q]};
                Cx t = cmul(dp, cmul(Mh[p][q], dq));
                S[p][q] = t.r;
            }
        float* o = ws + jz*12;
        o[0] = S[0][0]; o[1] = S[1][1]; o[2] = S[2][2]; o[3] = S[3][3];
        o[4] = 2.f*S[0][1]; o[5] = 2.f*S[0][2]; o[6] = 2.f*S[0][3];
        o[7] = 2.f*S[1][2]; o[8] = 2.f*S[1][3]; o[9] = 2.f*S[2][3];
        o[10] = 0.f; o[11] = 0.f;
    }
}

// ---------------------------------------------------------------------------
// Main kernel: per wave, 16 samples per step; features y[0..9] form the
// B-matrix (4x16 f32 per chunk), constant weight rows form A (16x4 f32),
// three chained V_WMMA_F32_16X16X4_F32 accumulate z0,z1 in full f32.
// ---------------------------------------------------------------------------
#define QITERS 4
#define PI_HALF 1.5707963267948966f

__global__ __launch_bounds__(256) void qcirc_wmma(const float* __restrict__ noise,
                                                  const float* __restrict__ W,
                                                  float* __restrict__ out,
                                                  int nsamp) {
    const int lane = threadIdx.x & 31;
    const int wave = (blockIdx.x * 256 + threadIdx.x) >> 5;
    const bool hi  = lane >= 16;
    const int  m   = lane & 15;
    const int  koff = hi ? 2 : 0;

    const int base0 = wave * (QITERS * 16);
    if (base0 >= nsamp) return;

    // A-matrix chunks (loop-invariant): rows 0,1 = weight rows for z0,z1.
    v2f a0 = {0.f, 0.f}, a1 = {0.f, 0.f}, a2 = {0.f, 0.f};
    if (m < 2) {
        const float* wr = W + m*12;
        a0.x = wr[0 + koff]; a0.y = wr[1 + koff];
        a1.x = wr[4 + koff]; a1.y = wr[5 + koff];
        a2.x = wr[8 + koff]; a2.y = wr[9 + koff];
    }

    for (int it = 0; it < QITERS; ++it) {
        const int s = base0 + it*16 + m;     // this lane's sample (dup'd in hi half)
        const float2 nz = reinterpret_cast<const float2*>(noise)[s];

        const float c0 = __cosf(nz.x * PI_HALF);
        const float s0 = __sinf(nz.x * PI_HALF);
        const float c1 = __cosf(nz.y * PI_HALF);
        const float s1 = __sinf(nz.y * PI_HALF);
        const float x0 = c0*c1, x1 = c0*s1, x2 = s0*c1, x3 = s0*s1;

        // Feature chunks: K = {y0..y3}, {y4..y7}, {y8,y9,0,0}
        v2f b0, b1, b2;
        b0.x = hi ? x2*x2 : x0*x0;  b0.y = hi ? x3*x3 : x1*x1;
        b1.x = hi ? x0*x3 : x0*x1;  b1.y = hi ? x1*x2 : x0*x2;
        b2.x = hi ? 0.f   : x1*x3;  b2.y = hi ? 0.f   : x2*x3;

        v8f c = {};
        c = __builtin_amdgcn_wmma_f32_16x16x4_f32(false, a0, false, b0, (short)0, c, false, false);
        c = __builtin_amdgcn_wmma_f32_16x16x4_f32(false, a1, false, b1, (short)0, c, false, false);
        c = __builtin_amdgcn_wmma_f32_16x16x4_f32(false, a2, false, b2, (short)0, c, false, false);

        // D layout: VGPR r, lanes 0-15 -> row M=r, col N=lane.
        if (!hi) {
            float2 r;
            r.x = c[0];   // z0 for sample s
            r.y = c[1];   // z1 for sample s
            reinterpret_cast<float2*>(out)[s] = r;
        }
    }
}

extern "C" void kernel_launch(void* const* d_in, const int* in_sizes, int n_in,
                              void* d_out, int out_size, void* d_ws, size_t ws_size,
                              hipStream_t stream) {
    const float* noise = (const float*)d_in[0];   // [B,2] f32
    const float* qw    = (const float*)d_in[1];   // [2,2,3] f32
    float* out = (float*)d_out;                   // [B,2] f32
    float* ws  = (float*)d_ws;                    // >= 24 floats

    precompute_weights<<<1, 32, 0, stream>>>(qw, ws);

    const int nsamp = in_sizes[0] / 2;            // 2,097,152
    const int samples_per_block = 8 * QITERS * 16;  // 8 waves/block
    const int blocks = (nsamp + samples_per_block - 1) / samples_per_block;
    qcirc_wmma<<<blocks, 256, 0, stream>>>(noise, ws, out, nsamp);
}